// LIFLayer_14422500180577
// MI455X (gfx1250) — compile-verified
//
#include <hip/hip_runtime.h>
#include <stdint.h>

// ---------------------------------------------------------------------------
// LIF layer fused kernel for gfx1250 (MI455X):
//   I = S @ W^T + b  (f32 WMMA 16x16x4), then sequential leaky-integrate-fire
//   scan over T with membrane state held in registers.
//
// Tiling: grid = (B/16, OUT/64); block = 128 threads = 4 waves.
//   wave w computes the 16x16 (M x N) tile at columns n_base + 16*w.
//   W tile (64 x 1024 f32, 256 KB) -> LDS once via TDM, reused for all T.
//   A tile (16 x 1024 f32,  64 KB) -> LDS per timestep via TDM.
// ---------------------------------------------------------------------------

typedef float v2f __attribute__((ext_vector_type(2)));
typedef float v8f __attribute__((ext_vector_type(8)));
typedef unsigned int v4u __attribute__((ext_vector_type(4)));
typedef int v4i __attribute__((ext_vector_type(4)));
typedef int v8i __attribute__((ext_vector_type(8)));

#define T_STEPS 100
#define BATCH   256
#define IN_F    1024
#define OUT_F   1024
#define M_WG    16
#define N_WG    64
#define TAU_F   20.0f
#define V_TH_F  1.0f

// Issue one 2D TDM tile load: global (tile_d1 rows x tile_d0 elems, f32,
// row stride stride_d0 elems) -> LDS at byte offset lds_off (compact).
__device__ __forceinline__ void tdm_load_2d_f32(uint32_t lds_off,
                                                const void* gptr,
                                                uint32_t tile_d0, uint32_t tile_d1,
                                                uint32_t stride_d0,
                                                uint32_t tensor_d0, uint32_t tensor_d1) {
  uint64_t ga = (uint64_t)(uintptr_t)gptr;
  v4u g0;
  g0[0] = 1u;                                   // count=1, user descriptor
  g0[1] = lds_off;                              // lds_addr (bytes)
  g0[2] = (uint32_t)(ga & 0xFFFFFFFFu);         // global_addr[31:0]  (bits 95:64)
  g0[3] = (uint32_t)((ga >> 32) & 0x01FFFFFFu)  // global_addr[56:32] (bits 120:96)
        | (2u << 30);                           // type = 2 ("image")
  v8i g1;
  g1[0] = (int)(2u << 16);                      // data_size=2 (4B); no multicast/pad/iter
  g1[1] = (int)((tensor_d0 & 0xFFFFu) << 16);   // tensor_dim0[15:0] @ bits 63:48
  g1[2] = (int)(((tensor_d0 >> 16) & 0xFFFFu) |
                ((tensor_d1 & 0xFFFFu) << 16)); // dim0[31:16] | dim1[15:0]
  g1[3] = (int)(((tensor_d1 >> 16) & 0xFFFFu) |
                ((tile_d0 & 0xFFFFu) << 16));   // dim1[31:16] | tile_dim0
  g1[4] = (int)(tile_d1 & 0xFFFFu);             // tile_dim1 (tile_dim2 = 0)
  g1[5] = (int)stride_d0;                       // tensor_dim0_stride[31:0]
  g1[6] = 0;                                    // stride0 hi | dim1_stride lo
  g1[7] = 0;                                    // dim1_stride hi
  v4i z4 = {0, 0, 0, 0};                        // groups 2/3: dims unused (2D tile)
  v8i z8 = {0, 0, 0, 0, 0, 0, 0, 0};            // group 4 (clang-23 form): unused
  __builtin_amdgcn_tensor_load_to_lds(g0, g1, z4, z4, z8, 0);
}

__global__ __launch_bounds__(128)
void lif_fused_wmma(const float* __restrict__ S,     // [T, B, IN]
                    const float* __restrict__ W,     // [OUT, IN]
                    const float* __restrict__ bias,  // [OUT]
                    float* __restrict__ out) {       // [T, B, OUT]
  __shared__ float Wl[N_WG * IN_F];   // 256 KB: W tile, n-major (resident all T)
  __shared__ float Al[M_WG * IN_F];   // 64 KB : spike tile for current t

  const int tid  = threadIdx.x;
  const int wave = tid >> 5;
  const int lane = tid & 31;
  const int h    = lane >> 4;   // lane half: selects K pair (A/B) and M half (C)
  const int nl   = lane & 15;

  const int m_base = blockIdx.x * M_WG;
  const int n_base = blockIdx.y * N_WG;
  const int n      = n_base + wave * 16 + nl;

  // Stage the W tile once with the Tensor Data Mover (wave 0 issues the DMA).
  if (wave == 0) {
    tdm_load_2d_f32((uint32_t)(size_t)&Wl[0],
                    W + (size_t)n_base * IN_F,
                    IN_F, N_WG, IN_F, IN_F, OUT_F);
    __builtin_amdgcn_s_wait_tensorcnt(0);
  }
  __syncthreads();

  const float bn   = bias[n];
  const float leak = 1.0f - 1.0f / TAU_F;

  float v[8];
#pragma unroll
  for (int r = 0; r < 8; ++r) v[r] = 0.0f;

  // Per-lane fragment base pointers (ISA 7.12.2 layouts):
  //   A 16x4 f32: lane holds row M=nl, K = k0 + 2*h + {0,1}  -> consecutive pair
  //   B  4x16 f32: lane holds col N=nl, K = k0 + 2*h + {0,1} -> consecutive pair
  const float* Alane = &Al[(size_t)nl * IN_F + 2 * h];
  const float* Wlane = &Wl[((size_t)wave * 16 + nl) * IN_F + 2 * h];

  for (int t = 0; t < T_STEPS; ++t) {
    // DMA this timestep's 16 x 1024 spike tile into LDS.
    if (wave == 0) {
      tdm_load_2d_f32((uint32_t)(size_t)&Al[0],
                      S + ((size_t)t * BATCH + m_base) * IN_F,
                      IN_F, M_WG, IN_F, IN_F, BATCH);
      __builtin_amdgcn_s_wait_tensorcnt(0);
    }
    __syncthreads();

    // I tile = A(16x1024) x B(1024x16), f32 WMMA, K-steps of 4.
    v8f c = {};
#pragma unroll 8
    for (int k0 = 0; k0 < IN_F; k0 += 4) {
      v2f a = *(const v2f*)(Alane + k0);
      v2f b = *(const v2f*)(Wlane + k0);
      c = __builtin_amdgcn_wmma_f32_16x16x4_f32(
              /*neg_a=*/false, a, /*neg_b=*/false, b,
              /*c_mod=*/(short)0, c, /*reuse_a=*/false, /*reuse_b=*/false);
    }

    // LIF update + spike store. C/D layout: VGPR r -> M = r + 8*h, N = nl.
    float* orow = out + ((size_t)t * BATCH + (size_t)(m_base + 8 * h)) * OUT_F + n;
#pragma unroll
    for (int r = 0; r < 8; ++r) {
      float I  = c[r] + bn;
      float vr = v[r] * leak + I;            // v += (-v/tau + I)*dt, dt=1
      float sp = (vr > V_TH_F) ? 1.0f : 0.0f;
      orow[(size_t)r * OUT_F] = sp;
      v[r] = (vr > V_TH_F) ? 0.0f : vr;      // reset on spike
    }
    __syncthreads();   // all waves done reading Al before next DMA overwrites it
  }
}

extern "C" void kernel_launch(void* const* d_in, const int* in_sizes, int n_in,
                              void* d_out, int out_size, void* d_ws, size_t ws_size,
                              hipStream_t stream) {
  (void)in_sizes; (void)n_in; (void)d_ws; (void)ws_size; (void)out_size;
  const float* S  = (const float*)d_in[0];   // [100, 256, 1024]
  const float* W  = (const float*)d_in[1];   // [1024, 1024]
  const float* b  = (const float*)d_in[2];   // [1024]
  float* out      = (float*)d_out;           // [100, 256, 1024]

  dim3 grid(BATCH / M_WG, OUT_F / N_WG);     // 16 x 16 = 256 workgroups
  lif_fused_wmma<<<grid, 128, 0, stream>>>(S, W, b, out);
}